// DeltaNet_31877247271450
// MI455X (gfx1250) — compile-verified
//
#include <hip/hip_runtime.h>
#include <hip/hip_bf16.h>

// DeltaNet chunkwise forward for gfx1250 (MI455X), wave32 + V_WMMA_F32_16X16X4_F32
// + async global->LDS DMA (GLOBAL_LOAD_ASYNC_TO_LDS_B128 / ASYNCcnt) for the v tile.
// Grid: one workgroup per (b,h,dv-slice); S-slice (256x64 f32) lives in LDS.
// LDS strides padded for 64-bank conflict-free WMMA fragment loads.

typedef float v2f __attribute__((ext_vector_type(2)));
typedef float v8f __attribute__((ext_vector_type(8)));

#define B_   2
#define H_   4
#define L_   4096
#define DK_  256
#define DV_  256
#define C_   32
#define DVS_ 64            // dv slice per workgroup
#define NC_  (L_ / C_)     // 128 chunks
#define NSL_ (DV_ / DVS_)  // 4 slices
#define OTOT_ ((size_t)B_ * H_ * L_ * DV_)  // o elements before S in d_out

// padded LDS strides (dwords): chosen so fragment accesses spread across 64 banks
#define DKP_  276   // 32x256 arrays: 276%64=20 -> A-row frags conflict-free; 8*276%64=32 -> C/D halves disjoint
#define DVP_  80    // 32x64 + S arrays: 80%64=16 -> B k-row frags tile all 64 banks
#define CP_   36    // 32x32 arrays: 36*sub mod 64 is a 16-permutation; 8*36%64=32

#define SMEM_FLOATS (DK_*DVP_ + 4*C_*DKP_ + 3*C_*DVP_ + 4*C_*CP_ + 512 + 64 + 32)
#define SMEM_BYTES  (SMEM_FLOATS * 4)

__device__ __forceinline__ v8f zero8() {
  v8f c; c[0]=0.f;c[1]=0.f;c[2]=0.f;c[3]=0.f;c[4]=0.f;c[5]=0.f;c[6]=0.f;c[7]=0.f;
  return c;
}

__device__ __forceinline__ v8f wmma4(v2f a, v2f b, v8f c) {
  return __builtin_amdgcn_wmma_f32_16x16x4_f32(false, a, false, b, (short)0, c, false, false);
}

// async DMA: 16 bytes per lane, global -> LDS; tracked by ASYNCcnt
__device__ __forceinline__ void async_load_b128(uint32_t lds_off, const void* gaddr) {
  asm volatile("global_load_async_to_lds_b128 %0, %1, off"
               :: "v"(lds_off), "v"(gaddr) : "memory");
}
__device__ __forceinline__ void wait_async0() {
  asm volatile("s_wait_asynccnt 0x0" ::: "memory");
}

// ---- generic K-loop tile GEMM: C += A(16xK) @ B(Kx16) ----
__device__ __forceinline__ v8f gemm_tile(const float* __restrict__ A, int a_rs, int a_cs,
                                         const float* __restrict__ Bm, int b_rs, int b_cs,
                                         int K, v8f c) {
  const int lane = threadIdx.x & 31;
  const int half = lane >> 4, sub = lane & 15;
  const float* ap = A  + sub * a_rs + (half * 2) * a_cs;
  const float* bp = Bm + (half * 2) * b_rs + sub * b_cs;
  for (int k0 = 0; k0 < K; k0 += 4) {
    v2f a, b;
    a.x = ap[0]; a.y = ap[a_cs];
    b.x = bp[0]; b.y = bp[b_rs];
    c = wmma4(a, b, c);
    ap += 4 * a_cs;
    bp += 4 * b_rs;
  }
  return c;
}

// ---- dual-A shared-B K-loop (B fragment loaded once, two wmmas) ----
__device__ __forceinline__ void gemm_dualA(const float* __restrict__ A0,
                                           const float* __restrict__ A1, int a_rs, int a_cs,
                                           const float* __restrict__ Bm, int b_rs, int b_cs,
                                           int K, v8f& c0, v8f& c1) {
  const int lane = threadIdx.x & 31;
  const int half = lane >> 4, sub = lane & 15;
  const float* a0p = A0 + sub * a_rs + (half * 2) * a_cs;
  const float* a1p = A1 + sub * a_rs + (half * 2) * a_cs;
  const float* bp  = Bm + (half * 2) * b_rs + sub * b_cs;
  for (int k0 = 0; k0 < K; k0 += 4) {
    v2f b;  b.x = bp[0];  b.y = bp[b_rs];
    v2f a0; a0.x = a0p[0]; a0.y = a0p[a_cs];
    v2f a1; a1.x = a1p[0]; a1.y = a1p[a_cs];
    c0 = wmma4(a0, b, c0);
    c1 = wmma4(a1, b, c1);
    a0p += 4 * a_cs; a1p += 4 * a_cs; bp += 4 * b_rs;
  }
}

// ---- register-resident A fragment for K=32 GEMMs (16 floats/lane) ----
struct AF32 { v2f a[8]; };

__device__ __forceinline__ AF32 load_afrag32(const float* __restrict__ A, int a_rs, int a_cs) {
  const int lane = threadIdx.x & 31;
  const int half = lane >> 4, sub = lane & 15;
  const float* ap = A + sub * a_rs + (half * 2) * a_cs;
  AF32 f;
#pragma unroll
  for (int s = 0; s < 8; ++s) {
    f.a[s].x = ap[0]; f.a[s].y = ap[a_cs];
    ap += 4 * a_cs;
  }
  return f;
}

__device__ __forceinline__ v8f gemm_b32(const AF32& fa, const float* __restrict__ Bm,
                                        int b_rs, int b_cs, v8f c) {
  const int lane = threadIdx.x & 31;
  const int half = lane >> 4, sub = lane & 15;
  const float* bp = Bm + (half * 2) * b_rs + sub * b_cs;
#pragma unroll
  for (int s = 0; s < 8; ++s) {
    v2f b; b.x = bp[0]; b.y = bp[b_rs];
    c = wmma4(fa.a[s], b, c);
    bp += 4 * b_rs;
  }
  return c;
}

// dual-A, shared-B, K=32 (both A fragments cached in registers)
__device__ __forceinline__ void gemm_b32_dual(const AF32& fa0, const AF32& fa1,
                                              const float* __restrict__ Bm, int b_rs, int b_cs,
                                              v8f& c0, v8f& c1) {
  const int lane = threadIdx.x & 31;
  const int half = lane >> 4, sub = lane & 15;
  const float* bp = Bm + (half * 2) * b_rs + sub * b_cs;
#pragma unroll
  for (int s = 0; s < 8; ++s) {
    v2f b; b.x = bp[0]; b.y = bp[b_rs];
    c0 = wmma4(fa0.a[s], b, c0);
    c1 = wmma4(fa1.a[s], b, c1);
    bp += 4 * b_rs;
  }
}

__device__ __forceinline__ void store_tile(float* dst, int ldd, v8f c) {
  const int lane = threadIdx.x & 31;
  const int half = lane >> 4, sub = lane & 15;
#pragma unroll
  for (int g = 0; g < 8; ++g) dst[(g + half * 8) * ldd + sub] = c[g];
}

__device__ __forceinline__ v8f load_tile(const float* src, int ldd) {
  const int lane = threadIdx.x & 31;
  const int half = lane >> 4, sub = lane & 15;
  v8f c;
#pragma unroll
  for (int g = 0; g < 8; ++g) c[g] = src[(g + half * 8) * ldd + sub];
  return c;
}

// store with inclusive-lower-triangular mask (keep col <= row in global 32x32 coords)
__device__ __forceinline__ void store_tile_tril(float* dst, int ldd, v8f c, int mBase, int nBase) {
  const int lane = threadIdx.x & 31;
  const int half = lane >> 4, sub = lane & 15;
#pragma unroll
  for (int g = 0; g < 8; ++g) {
    const int m = g + half * 8;
    dst[m * ldd + sub] = ((nBase + sub) <= (mBase + m)) ? c[g] : 0.f;
  }
}

__global__ void __launch_bounds__(256, 1)
deltanet_chunk_scan(const float* __restrict__ q, const float* __restrict__ k,
                    const float* __restrict__ v, const float* __restrict__ beta,
                    float* __restrict__ out) {
  extern __shared__ float smem[];
  float* S    = smem;                // 256 x DVP_ (state slice, padded)
  float* Qc   = S    + DK_ * DVP_;   // 32 x DKP_ (l2-normalized q chunk)
  float* Kc   = Qc   + C_ * DKP_;    // 32 x DKP_ (l2-normalized k chunk)
  float* KB   = Kc   + C_ * DKP_;    // 32 x DKP_ (k * beta)
  float* W    = KB   + C_ * DKP_;    // 32 x DKP_ (T @ k_beta)
  float* Vc   = W    + C_ * DKP_;    // 32 x DVP_ (RAW v slice, async-DMAed)
  float* U    = Vc   + C_ * DVP_;    // 32 x DVP_ (T_beta @ v)
  float* UH   = U    + C_ * DVP_;    // 32 x DVP_ (u - w@S)
  float* Pm   = UH   + C_ * DVP_;    // 32 x CP_  (k_beta @ k^T)
  float* Tm   = Pm   + C_ * CP_;     // 32 x CP_  ((I - tril_strict(-P))^{-1})
  float* TB   = Tm   + C_ * CP_;     // 32 x CP_  (T with beta folded into columns)
  float* ATT  = TB   + C_ * CP_;     // 32 x CP_  (tril(q @ k^T))
  float* PS   = ATT  + C_ * CP_;     // 512 reduction partials
  float* RN   = PS   + 512;          // 64 row rsqrt norms (q then k)
  float* BETA = RN   + 64;           // 32 per-row beta

  const int tid = threadIdx.x;
  const int wv  = tid >> 5;          // wave id, 0..7
  const int bh  = blockIdx.x >> 2;   // (b*H + h), 0..7
  const int sl  = blockIdx.x & 3;    // dv slice
  const int dvbase = sl * DVS_;

  // zero state slice
  for (int i = tid; i < DK_ * DVP_; i += 256) S[i] = 0.f;
  __syncthreads();

  const int r  = tid >> 3;           // row 0..31 handled by this thread in load phase
  const int c8 = tid & 7;            // 8 threads per row

  for (int n = 0; n < NC_; ++n) {
    const int l0 = n * C_;

    // ---- kick off async DMA of the raw v tile (32 x 64) into LDS:
    //      2 x b128 per lane (512B per wave-issue); overlaps with q/k normalization ----
#pragma unroll
    for (int t = 0; t < 2; ++t) {
      const int s    = tid + t * 256;     // 512 x 16B segments
      const int row  = s >> 4;
      const int col4 = (s & 15) * 4;
      const uint32_t ldsoff = (uint32_t)(uintptr_t)(Vc + row * DVP_ + col4);
      const float* gp = v + ((size_t)bh * L_ + l0 + row) * DV_ + dvbase + col4;
      async_load_b128(ldsoff, gp);
    }

    // ---- load chunk, l2-normalize q/k rows, apply beta ----
    {
      const size_t rowoff = (size_t)bh * L_ + l0 + r;
      const float* qrow = q + rowoff * DK_ + c8 * 32;
      const float* krow = k + rowoff * DK_ + c8 * 32;
      float qv[32], kv[32];
      float sq = 0.f, sk = 0.f;
#pragma unroll
      for (int i = 0; i < 32; i += 4) {
        const float4 qa = *reinterpret_cast<const float4*>(qrow + i);
        const float4 ka = *reinterpret_cast<const float4*>(krow + i);
        qv[i] = qa.x; qv[i+1] = qa.y; qv[i+2] = qa.z; qv[i+3] = qa.w;
        kv[i] = ka.x; kv[i+1] = ka.y; kv[i+2] = ka.z; kv[i+3] = ka.w;
        sq += qa.x*qa.x + qa.y*qa.y + qa.z*qa.z + qa.w*qa.w;
        sk += ka.x*ka.x + ka.y*ka.y + ka.z*ka.z + ka.w*ka.w;
      }
      PS[r * 8 + c8]       = sq;
      PS[256 + r * 8 + c8] = sk;
      if (n + 1 < NC_) {  // prefetch next chunk rows (global_prefetch_b8)
        __builtin_prefetch(qrow + (size_t)C_ * DK_, 0, 1);
        __builtin_prefetch(krow + (size_t)C_ * DK_, 0, 1);
      }
      __syncthreads();
      if (tid < 64) {
        const int rr = tid & 31, which = tid >> 5;
        const float* p = PS + which * 256 + rr * 8;
        RN[which * 32 + rr] =
            rsqrtf(p[0] + p[1] + p[2] + p[3] + p[4] + p[5] + p[6] + p[7]);
      }
      __syncthreads();
      const float bet = beta[(size_t)bh * L_ + l0 + r];
      if (c8 == 0) BETA[r] = bet;
      const float rq = RN[r], rk = RN[32 + r];
      float* qd  = Qc + r * DKP_ + c8 * 32;
      float* kd  = Kc + r * DKP_ + c8 * 32;
      float* kbd = KB + r * DKP_ + c8 * 32;
#pragma unroll
      for (int i = 0; i < 32; ++i) {
        qd[i] = qv[i] * rq;
        const float kn = kv[i] * rk;
        kd[i]  = kn;
        kbd[i] = kn * bet;
      }
    }
    wait_async0();   // v tile resident in LDS (this wave's DMAs done)
    __syncthreads(); // publish Qc/Kc/KB/Vc/BETA to all waves

    // ---- P = k_beta @ k^T (waves 0-3) ; ATT = tril(q @ k^T) (waves 4-7); K=256 ----
    if (wv < 4) {
      const int mi = wv >> 1, ni = wv & 1;
      v8f c = zero8();
      c = gemm_tile(KB + mi * 16 * DKP_, DKP_, 1, Kc + ni * 16 * DKP_, 1, DKP_, DK_, c);
      store_tile(Pm + mi * 16 * CP_ + ni * 16, CP_, c);
    } else {
      const int w4 = wv - 4;
      const int mi = w4 >> 1, ni = w4 & 1;
      v8f c = zero8();
      c = gemm_tile(Qc + mi * 16 * DKP_, DKP_, 1, Kc + ni * 16 * DKP_, 1, DKP_, DK_, c);
      store_tile_tril(ATT + mi * 16 * CP_ + ni * 16, CP_, c, mi * 16, ni * 16);
    }
    __syncthreads();

    // ---- T = (I - tril_strict(-P))^{-1}: forward substitution, wave 0, lane=column.
    //      TB = T * diag(beta) (column-scaled) so U can use the raw v tile. ----
    if (wv == 0) {
      const int lane = tid & 31;
      const float bcol = BETA[lane];
      for (int i = 0; i < C_; ++i) {
        float t = (lane == i) ? 1.f : 0.f;
        for (int j = 0; j < i; ++j) t -= Pm[i * CP_ + j] * Tm[j * CP_ + lane];
        Tm[i * CP_ + lane] = t;        // lane only re-reads its own column
        TB[i * CP_ + lane] = t * bcol;
      }
    }
    __syncthreads();

    // ---- W = T @ k_beta (4 tiles/wave) ; U = (T*beta) @ v_raw (1 tile/wave); K=32,
    //      T row-block fragments cached in registers ----
    {
      const int mi = wv >> 2;           // same row block for this wave's W and U tiles
      const AF32 ft = load_afrag32(Tm + mi * 16 * CP_, CP_, 1);
#pragma unroll
      for (int t = 0; t < 4; ++t) {
        const int ni = (wv & 3) * 4 + t;
        v8f c = zero8();
        c = gemm_b32(ft, KB + ni * 16, DKP_, 1, c);
        store_tile(W + mi * 16 * DKP_ + ni * 16, DKP_, c);
      }
      {
        const AF32 ftb = load_afrag32(TB + mi * 16 * CP_, CP_, 1);
        const int ni = wv & 3;
        v8f c = zero8();
        c = gemm_b32(ftb, Vc + ni * 16, DVP_, 1, c);
        store_tile(U + mi * 16 * DVP_ + ni * 16, DVP_, c);
      }
    }
    __syncthreads();

    // ---- u_hat = U - W @ S ; qS = Q @ S : one 16x16 tile per wave, K=256,
    //      S fragment loaded once per k-step, shared by both accumulators ----
    const int mi = wv >> 2, ni = wv & 3;
    v8f cq = zero8();
    v8f cw = zero8();
    gemm_dualA(Qc + mi * 16 * DKP_, W + mi * 16 * DKP_, DKP_, 1,
               S + ni * 16, DVP_, 1, DK_, cq, cw);
    {
      const v8f uu = load_tile(U + mi * 16 * DVP_ + ni * 16, DVP_);
      store_tile(UH + mi * 16 * DVP_ + ni * 16, DVP_, uu - cw);
    }
    __syncthreads();

    // ---- o = qS + ATT @ u_hat -> global ; K=32, ATT fragment in registers ----
    {
      const AF32 fat = load_afrag32(ATT + mi * 16 * CP_, CP_, 1);
      cq = gemm_b32(fat, UH + ni * 16, DVP_, 1, cq);
      const int lane = tid & 31, half = lane >> 4, sub = lane & 15;
      float* ob = out + ((size_t)bh * L_ + l0 + mi * 16) * DV_ + dvbase + ni * 16;
#pragma unroll
      for (int g = 0; g < 8; ++g) ob[(size_t)(g + half * 8) * DV_ + sub] = cq[g];
    }

    // ---- S += k^T @ u_hat : 256x64; wave owns row blocks {2wv, 2wv+1} x 4 cols; K=32.
    //      Both k^T fragments cached in registers; u_hat B fragment shared. ----
    {
      const AF32 fk0 = load_afrag32(Kc + (2 * wv + 0) * 16, 1, DKP_);
      const AF32 fk1 = load_afrag32(Kc + (2 * wv + 1) * 16, 1, DKP_);
#pragma unroll
      for (int sni = 0; sni < 4; ++sni) {
        float* sp0 = S + (2 * wv + 0) * 16 * DVP_ + sni * 16;
        float* sp1 = S + (2 * wv + 1) * 16 * DVP_ + sni * 16;
        v8f c0 = load_tile(sp0, DVP_);
        v8f c1 = load_tile(sp1, DVP_);
        gemm_b32_dual(fk0, fk1, UH + sni * 16, DVP_, 1, c0, c1);
        store_tile(sp0, DVP_, c0);
        store_tile(sp1, DVP_, c1);
      }
    }
    __syncthreads();  // protect S / Kc / UH / Vc before next chunk's load+DMA phase
  }

  // ---- write final state slice: d_out[OTOT_ + bh*DK*DV + row*DV + dvbase + col] ----
  for (int i = tid; i < DK_ * DVS_; i += 256) {
    const int rr = i / DVS_, cc = i % DVS_;
    out[OTOT_ + ((size_t)bh * DK_ + rr) * DV_ + dvbase + cc] = S[rr * DVP_ + cc];
  }
}

extern "C" void kernel_launch(void* const* d_in, const int* in_sizes, int n_in,
                              void* d_out, int out_size, void* d_ws, size_t ws_size,
                              hipStream_t stream) {
  const float* q    = (const float*)d_in[0];
  const float* k    = (const float*)d_in[1];
  const float* v    = (const float*)d_in[2];
  const float* beta = (const float*)d_in[3];
  float* out        = (float*)d_out;

  (void)in_sizes; (void)n_in; (void)out_size; (void)d_ws; (void)ws_size;

  // allow >64KB dynamic LDS (~275 KB used; WGP supports up to 320 KB)
  hipFuncSetAttribute(reinterpret_cast<const void*>(deltanet_chunk_scan),
                      hipFuncAttributeMaxDynamicSharedMemorySize, SMEM_BYTES);

  dim3 grid(B_ * H_ * NSL_);  // 32 workgroups: 8 heads x 4 dv-slices
  dim3 block(256);            // 8 wave32 waves
  deltanet_chunk_scan<<<grid, block, SMEM_BYTES, stream>>>(q, k, v, beta, out);
}